// FourierRingCorrelation_21320217658068
// MI455X (gfx1250) — compile-verified
//
#include <hip/hip_runtime.h>

// ---------------------------------------------------------------------------
// FRC via GEMM-formulated 2D DFT on CDNA5 (gfx1250), wave32.
//   F = Wdft * (A * Wdft),  Wdft[j][k] = exp(-2*pi*i*j*k/N) (symmetric).
// fp32 matrix path (V_WMMA_F32_16X16X4_F32): ~206 GFLOP total, ~0.5 GB HBM
// traffic (~20 us at 23.3 TB/s) -> bound by fp32 matrix issue, far under
// chip limits.  fp32 twiddles keep the 1024-point reductions accurate.
//
// Data movement: B-operand panels (32x64 fp32 tiles) are staged into LDS with
// the Tensor Data Mover (tensor_load_to_lds, TENSORcnt), double-buffered and
// issued by wave 0 while all 8 waves run WMMA on the current panel — the
// canonical CDNA5 async-GEMM pipeline.  A-operands stay as per-lane global
// loads (row reuse hits WGP$/L2; the 8 MB twiddle tables are L2-resident).
// ---------------------------------------------------------------------------

typedef float v2f __attribute__((ext_vector_type(2)));
typedef float v8f __attribute__((ext_vector_type(8)));
typedef unsigned int v4u __attribute__((ext_vector_type(4)));
typedef unsigned int v8u __attribute__((ext_vector_type(8)));

constexpr int    NS    = 1024;          // H == W
constexpr int    NB    = 8;             // batches
constexpr int    NIMG  = 16;            // 8 X-images + 8 Y-images
constexpr int    NRING = 512;           // min(H,W) / (2*delta)
constexpr int    KC    = 32;            // K-chunk staged per TDM tile

// workspace layout, in floats
constexpr size_t OFF_C    = 0;                                   // cos table  (4 MiB)
constexpr size_t OFF_S    = (size_t)NS * NS;                     // sin table  (4 MiB)
constexpr size_t OFF_RI   = 2 * (size_t)NS * NS;                 // row-DFT Re/Im per image (128 MiB)
constexpr size_t OFF_SUMS = OFF_RI + (size_t)NIMG * 2 * NS * NS; // ring sums 3*8*512

// one fp32 16x16x4 WMMA step
__device__ __forceinline__ v8f wmma4(v2f a, v2f b, v8f c) {
  return __builtin_amdgcn_wmma_f32_16x16x4_f32(false, a, false, b, (short)0, c,
                                               false, false);
}

// ---- Tensor Data Mover: DMA one KC x 64 fp32 tile (row-major, tensor row
// stride 1024 elements) from global into LDS.  D# built per ISA 08_async:
//   group0: count=1 | lds_addr | global_addr(57b) | type=2
//   group1: data_size=4B, tensor_dim0/1=1024, tile_dim0=64, tile_dim1=KC,
//           tensor_dim0_stride=1024
__device__ __forceinline__ void tdm_load_tile(unsigned int lds_byte_off,
                                              const float* gptr) {
  const unsigned long long ga = (unsigned long long)(uintptr_t)gptr;
  const v4u g0 = {
      1u,                                                  // count=1
      lds_byte_off,                                        // LDS dest (bytes)
      (unsigned int)ga,                                    // global_addr[31:0]
      (unsigned int)((ga >> 32) & 0x1FFFFFFu) | (2u << 30) // addr[56:32]|type=2
  };
  const v8u g1 = {
      2u << 16,                       // data_size=2 (4 bytes)
      (unsigned int)NS << 16,         // tensor_dim0 low16 (1024)
      (unsigned int)NS << 16,         // tensor_dim0 hi | tensor_dim1 low16
      64u << 16,                      // tensor_dim1 hi | tile_dim0 = 64
      (unsigned int)KC,               // tile_dim1 = KC, tile_dim2 = 0
      (unsigned int)NS,               // tensor_dim0_stride low32 (1024)
      0u,                             // dim0_stride hi | dim1_stride low
      0u                              // dim1_stride hi
  };
  asm volatile("tensor_load_to_lds %0, %1" :: "s"(g0), "s"(g1) : "memory");
}

__device__ __forceinline__ unsigned int lds_off(const void* p) {
  // LDS flat aperture: addr[31:0] is the offset within the wave's allocation
  return (unsigned int)(uintptr_t)p;
}

// -------------------- twiddle tables: exact (j*k mod N) angles --------------
__global__ __launch_bounds__(256) void k_twiddle(float* __restrict__ ws) {
  const int idx = blockIdx.x * 256 + threadIdx.x;          // 4096 blocks -> 1M
  const int j = idx >> 10;
  const int k = idx & (NS - 1);
  const int m = (j * k) & (NS - 1);                        // exact mod: period N
  const float ang = (float)m * (-6.283185307179586f / (float)NS);
  float s, c;
  __sincosf(ang, &s, &c);
  ws[OFF_C + idx] = c;
  ws[OFF_S + idx] = s;
}

// -------------------- zero the ring accumulators ----------------------------
__global__ __launch_bounds__(256) void k_zero(float* __restrict__ ws) {
  const int idx = blockIdx.x * 256 + threadIdx.x;          // 48 blocks = 12288
  ws[OFF_SUMS + idx] = 0.0f;
}

// -------------------- stage 1: row DFT (real input) -------------------------
// Re1 = A*C ; Im1 = A*S.  8 waves/block on a 32x64 macro-tile; cos/sin B
// panels TDM-staged into double-buffered LDS.  grid = (512 tiles, 16 images).
__global__ __launch_bounds__(256) void k_rowdft(const float* __restrict__ X,
                                                const float* __restrict__ Y,
                                                float* __restrict__ ws) {
  __shared__ __align__(128) float sB[2][2][KC * 64];       // [buf][cos/sin] 32KB
  const int tid  = threadIdx.x;
  const int lane = tid & 31;
  const int w    = tid >> 5;
  const int bx   = blockIdx.x;
  const int img  = blockIdx.y;
  const int tm   = bx & 31;              // 32 macro-tiles in M
  const int tn   = bx >> 5;              // 16 macro-tiles in N
  const int m0   = tm * 32 + (w >> 2) * 16;
  const int nloc = (w & 3) * 16;         // N offset inside the 64-wide panel
  const int lm   = lane & 15;
  const int kb   = (lane >> 4) * 2;      // K = vgpr + 2*(lane>=16)

  const float* A  = (img < NB) ? (X + (size_t)img * NS * NS)
                               : (Y + (size_t)(img - NB) * NS * NS);
  const float* Cm = ws + OFF_C;
  const float* Sm = ws + OFF_S;
  const float* arow = A + (size_t)(m0 + lm) * NS;
  const int nmac = tn * 64;

  v8f are = {};
  v8f aim = {};

  // prologue: chunk 0 into buffer 0 (wave 0 issues; TENSORcnt is per-wave)
  if (tid < 32) {
    tdm_load_tile(lds_off(&sB[0][0][0]), Cm + nmac);
    tdm_load_tile(lds_off(&sB[0][1][0]), Sm + nmac);
    __builtin_amdgcn_s_wait_tensorcnt(0);
  }
  __syncthreads();

  for (int chunk = 0; chunk < NS / KC; ++chunk) {
    const int buf = chunk & 1;
    if (tid < 32 && chunk + 1 < NS / KC) {
      const size_t gk = (size_t)(chunk + 1) * KC * NS + nmac;
      tdm_load_tile(lds_off(&sB[buf ^ 1][0][0]), Cm + gk);
      tdm_load_tile(lds_off(&sB[buf ^ 1][1][0]), Sm + gk);
    }
    const float* bC = &sB[buf][0][0];
    const float* bS = &sB[buf][1][0];
    const int k0g = chunk * KC;
    __builtin_prefetch(arow + k0g + KC, 0, 1);
#pragma unroll
    for (int kk = 0; kk < KC; kk += 4) {
      const int ka = kk + kb;
      const v2f a = *(const v2f*)(arow + k0g + ka);        // A fragment (16x4)
      const int lo = ka * 64 + nloc + lm;
      v2f bc, bs;                                          // B fragments (4x16)
      bc.x = bC[lo]; bc.y = bC[lo + 64];
      bs.x = bS[lo]; bs.y = bS[lo + 64];
      are = wmma4(a, bc, are);
      aim = wmma4(a, bs, aim);
    }
    if (tid < 32 && chunk + 1 < NS / KC) __builtin_amdgcn_s_wait_tensorcnt(0);
    __syncthreads();
  }

  float* Re = ws + OFF_RI + (size_t)img * 2 * NS * NS;
  float* Im = Re + (size_t)NS * NS;
  const int rbase = m0 + ((lane >> 4) ? 8 : 0);            // M = vgpr + 8*(lane>=16)
  const int col   = nmac + nloc + lm;
#pragma unroll
  for (int v = 0; v < 8; ++v) {
    Re[(size_t)(rbase + v) * NS + col] = are[v];
    Im[(size_t)(rbase + v) * NS + col] = aim[v];
  }
}

// -------------------- stage 2: column DFT + ring accumulation ---------------
// F = Wc * R (complex) for X and Y together; Xre/Xim/Yre/Yim panels TDM-staged
// into double-buffered LDS; epilogue bins cross/p1/p2 into an LDS ring
// histogram and flushes with global fp32 atomics.  grid = (512 tiles, 8).
__global__ __launch_bounds__(256) void k_coldft(float* __restrict__ ws) {
  __shared__ __align__(128) float sB[2][4][KC * 64];       // [buf][xr,xi,yr,yi] 64KB
  __shared__ float sh[3 * NRING];                          // 6KB ring histogram
  const int tid  = threadIdx.x;
  const int lane = tid & 31;
  const int w    = tid >> 5;
  const int bx   = blockIdx.x;
  const int b    = blockIdx.y;
  const int tm   = bx & 31;
  const int tn   = bx >> 5;
  const int m0   = tm * 32 + (w >> 2) * 16;
  const int nloc = (w & 3) * 16;
  const int lm   = lane & 15;
  const int kb   = (lane >> 4) * 2;
  const int nmac = tn * 64;

  for (int i = tid; i < 3 * NRING; i += 256) sh[i] = 0.0f;

  const float* Cm  = ws + OFF_C;
  const float* Sm  = ws + OFF_S;
  const float* Xre = ws + OFF_RI + (size_t)b * 2 * NS * NS;
  const float* Xim = Xre + (size_t)NS * NS;
  const float* Yre = ws + OFF_RI + (size_t)(b + NB) * 2 * NS * NS;
  const float* Yim = Yre + (size_t)NS * NS;
  const float* crow = Cm + (size_t)(m0 + lm) * NS;
  const float* srow = Sm + (size_t)(m0 + lm) * NS;

  v8f xre = {}, xim = {}, yre = {}, yim = {};

  if (tid < 32) {
    tdm_load_tile(lds_off(&sB[0][0][0]), Xre + nmac);
    tdm_load_tile(lds_off(&sB[0][1][0]), Xim + nmac);
    tdm_load_tile(lds_off(&sB[0][2][0]), Yre + nmac);
    tdm_load_tile(lds_off(&sB[0][3][0]), Yim + nmac);
    __builtin_amdgcn_s_wait_tensorcnt(0);
  }
  __syncthreads();

  for (int chunk = 0; chunk < NS / KC; ++chunk) {
    const int buf = chunk & 1;
    if (tid < 32 && chunk + 1 < NS / KC) {
      const size_t gk = (size_t)(chunk + 1) * KC * NS + nmac;
      tdm_load_tile(lds_off(&sB[buf ^ 1][0][0]), Xre + gk);
      tdm_load_tile(lds_off(&sB[buf ^ 1][1][0]), Xim + gk);
      tdm_load_tile(lds_off(&sB[buf ^ 1][2][0]), Yre + gk);
      tdm_load_tile(lds_off(&sB[buf ^ 1][3][0]), Yim + gk);
    }
    const float* bXr = &sB[buf][0][0];
    const float* bXi = &sB[buf][1][0];
    const float* bYr = &sB[buf][2][0];
    const float* bYi = &sB[buf][3][0];
    const int k0g = chunk * KC;
    __builtin_prefetch(crow + k0g + KC, 0, 1);
#pragma unroll
    for (int kk = 0; kk < KC; kk += 4) {
      const int ka = kk + kb;
      const v2f ac  = *(const v2f*)(crow + k0g + ka);
      const v2f as  = *(const v2f*)(srow + k0g + ka);
      const v2f asn = -as;     // fp32 WMMA NEG is C-only; negate A in VALU

      const int lo = ka * 64 + nloc + lm;
      v2f bxr, bxi, byr, byi;
      bxr.x = bXr[lo]; bxr.y = bXr[lo + 64];
      bxi.x = bXi[lo]; bxi.y = bXi[lo + 64];
      byr.x = bYr[lo]; byr.y = bYr[lo + 64];
      byi.x = bYi[lo]; byi.y = bYi[lo + 64];

      // Re2 = C*Re1 - S*Im1 ; Im2 = C*Im1 + S*Re1
      xre = wmma4(ac,  bxr, xre);  xre = wmma4(asn, bxi, xre);
      xim = wmma4(ac,  bxi, xim);  xim = wmma4(as,  bxr, xim);
      yre = wmma4(ac,  byr, yre);  yre = wmma4(asn, byi, yre);
      yim = wmma4(ac,  byi, yim);  yim = wmma4(as,  byr, yim);
    }
    if (tid < 32 && chunk + 1 < NS / KC) __builtin_amdgcn_s_wait_tensorcnt(0);
    __syncthreads();
  }

  const int rbase = m0 + ((lane >> 4) ? 8 : 0);
  const int gn    = nmac + nloc + lm;
  const int fx    = (gn < NS / 2) ? gn : gn - NS;
  const int fx2   = fx * fx;
#pragma unroll
  for (int v = 0; v < 8; ++v) {
    const int gm = rbase + v;
    const int fy = (gm < NS / 2) ? gm : gm - NS;
    const int r2 = fy * fy + fx2;
    int ib = (int)sqrtf((float)r2);              // exact integer floor(sqrt)
    while ((ib + 1) * (ib + 1) <= r2) ++ib;
    while (ib * ib > r2) --ib;
    if (ib < NRING) {                            // ring 512 is discarded
      const float cr = xre[v] * yre[v] + xim[v] * yim[v];
      const float p1 = xre[v] * xre[v] + xim[v] * xim[v];
      const float p2 = yre[v] * yre[v] + yim[v] * yim[v];
      atomicAdd(&sh[ib],             cr);
      atomicAdd(&sh[NRING + ib],     p1);
      atomicAdd(&sh[2 * NRING + ib], p2);
    }
  }
  __syncthreads();

  float* gs = ws + OFF_SUMS;                     // [3][NB][NRING]
  for (int i = tid; i < 3 * NRING; i += 256) {
    const float val = sh[i];
    if (val != 0.0f) {
      const int c = i / NRING;
      const int r = i - c * NRING;
      atomicAdd(&gs[((size_t)c * NB + b) * NRING + r], val);
    }
  }
}

// -------------------- stage 3: frc = cs/sqrt(s1*s2+eps), mean ---------------
__global__ __launch_bounds__(512) void k_final(const float* __restrict__ ws,
                                               float* __restrict__ out) {
  __shared__ float red[512];
  const int t = threadIdx.x;                     // t = ring index
  const float* gs = ws + OFF_SUMS;
  float acc = 0.0f;
#pragma unroll
  for (int b = 0; b < NB; ++b) {
    const float cs = gs[((size_t)0 * NB + b) * NRING + t];
    const float s1 = gs[((size_t)1 * NB + b) * NRING + t];
    const float s2 = gs[((size_t)2 * NB + b) * NRING + t];
    acc += cs * rsqrtf(s1 * s2 + 1e-8f);
  }
  red[t] = acc;
  __syncthreads();
  for (int s = 256; s > 0; s >>= 1) {
    if (t < s) red[t] += red[t + s];
    __syncthreads();
  }
  if (t == 0) out[0] = red[0] / (float)(NB * NRING);
}

// ---------------------------------------------------------------------------
extern "C" void kernel_launch(void* const* d_in, const int* in_sizes, int n_in,
                              void* d_out, int out_size, void* d_ws, size_t ws_size,
                              hipStream_t stream) {
  const float* X = (const float*)d_in[0];
  const float* Y = (const float*)d_in[1];
  float* out = (float*)d_out;
  float* ws  = (float*)d_ws;

  k_twiddle<<<dim3(4096), 256, 0, stream>>>(ws);
  k_zero   <<<dim3(48),   256, 0, stream>>>(ws);
  k_rowdft <<<dim3(512, NIMG), 256, 0, stream>>>(X, Y, ws);
  k_coldft <<<dim3(512, NB),   256, 0, stream>>>(ws);
  k_final  <<<1, 512, 0, stream>>>(ws, out);
}